// DDSL_spec_15461882265824
// MI455X (gfx1250) — compile-verified
//
#include <hip/hip_runtime.h>

typedef __attribute__((ext_vector_type(2))) float v2f;
typedef __attribute__((ext_vector_type(8))) float v8f;

#define NVELEM 512
#define BATCH  2
#define FX     128
#define FY2    65
#define NFREQ  (FX * FY2)          // 8320
#define TILES_PER_B (NFREQ / 16)   // 520 wave-tiles of 16 freqs
#define BLOCKS_PER_B (TILES_PER_B / 8) // 65 blocks (8 waves each)
#define PI2 6.28318530717958647692f

// ---------------------------------------------------------------------------
// Kernel 1: per-element prep. elem[b*NV+e] = {v0x, v0y, v1x, v1y, C*D}
// ---------------------------------------------------------------------------
__global__ void ddsl_prep(const float* __restrict__ V, const int* __restrict__ E,
                          const float* __restrict__ D, float* __restrict__ elem) {
  int tid = blockIdx.x * blockDim.x + threadIdx.x;
  if (tid >= BATCH * NVELEM) return;
  int b = tid / NVELEM, e = tid % NVELEM;
  int i0 = E[(b * NVELEM + e) * 2 + 0];
  int i1 = E[(b * NVELEM + e) * 2 + 1];
  float v0x = V[(b * NVELEM + i0) * 2 + 0];
  float v0y = V[(b * NVELEM + i0) * 2 + 1];
  float v1x = V[(b * NVELEM + i1) * 2 + 0];
  float v1y = V[(b * NVELEM + i1) * 2 + 1];
  // C = det of edge vectors relative to origin vertex (third vertex = 0)
  float cd = (v0x * v1y - v0y * v1x) * D[b * NVELEM + e];
  float* p = elem + tid * 5;
  p[0] = v0x; p[1] = v0y; p[2] = v1x; p[3] = v1y; p[4] = cd;
}

// ---------------------------------------------------------------------------
// Kernel 2: main accumulation. Each wave32 owns 16 frequencies; WMMA
// f32 16x16x4 computes sig = P . omega for 16 points x 16 freqs per step.
// ---------------------------------------------------------------------------
__global__ void __launch_bounds__(256) ddsl_main(const float* __restrict__ elem,
                                                 float* __restrict__ out) {
  __shared__ float lds[NVELEM * 5];   // 10240 B of the 320 KB WGP LDS
  const int b  = blockIdx.x / BLOCKS_PER_B;
  const int tb = blockIdx.x % BLOCKS_PER_B;

  for (int i = threadIdx.x; i < NVELEM * 5; i += 256)
    lds[i] = elem[b * NVELEM * 5 + i];
  __syncthreads();

  const int lane   = threadIdx.x & 31;
  const int laneLo = lane & 15;
  const int hi     = lane >> 4;          // 0: rows 0-7 / K0-1, 1: rows 8-15 / K2-3
  const int t      = tb * 8 + (threadIdx.x >> 5);
  const int f      = t * 16 + laneLo;    // flat frequency id, < 8320 always
  const int x      = f / FY2;
  const int y      = f % FY2;
  const float kx   = (x < 64) ? (float)x : (float)(x - 128);

  // B operand: 4x16, lanes 0-15 carry K rows 0..1 = (wx, wy); K rows 2..3 = 0
  v2f bm;
  bm.x = hi ? 0.0f : PI2 * kx;
  bm.y = hi ? 0.0f : PI2 * (float)y;

  float accRe = 0.0f, accIm = 0.0f;

  for (int pt = 0; pt < NVELEM / 8; ++pt) {
    const int elemBase = pt * 8;

    // A operand: 16 points (8 elements x {v0,v1}) by K=4 ([px,py,0,0])
    const int eA   = elemBase + (laneLo >> 1);
    const int vert = laneLo & 1;
    const float ax = lds[eA * 5 + vert * 2 + 0];
    const float ay = lds[eA * 5 + vert * 2 + 1];
    v2f am;
    am.x = hi ? 0.0f : ax;
    am.y = hi ? 0.0f : ay;

    v8f c = {};
    // sig tile: D[v] = phase of point row (hi ? 8+v : v) at freq column laneLo
    v8f sig = __builtin_amdgcn_wmma_f32_16x16x4_f32(
        false, am, false, bm, (short)0, c, false, false);

#pragma unroll
    for (int k = 0; k < 4; ++k) {
      const float s0 = sig[2 * k + 0];   // phase of v0, element ei
      const float s1 = sig[2 * k + 1];   // phase of v1, element ei
      const int   ei = elemBase + hi * 4 + k;
      const float cd = lds[ei * 5 + 4];

      const float d01  = s0 - s1;
      const float prod = s0 * s1 * d01;  // common denominator
      const float mask = (prod != 0.0f) ? 1.0f : 0.0f;
      const float rc   = mask * __builtin_amdgcn_rcpf((prod != 0.0f) ? prod : 1.0f);

      const float c0 = __cosf(s0), si0 = __sinf(s0);
      const float c1 = __cosf(s1), si1 = __sinf(s1);

      // cos(s0)/(s0*d01) + cos(s1)/(-s1*d01) + 1/(s0*s1), over common denom
      const float numRe = c0 * s1 - c1 * s0 + d01;
      const float numIm = si1 * s0 - si0 * s1;

      const float w = cd * rc;
      accRe = fmaf(w, numRe, accRe);
      accIm = fmaf(w, numIm, accIm);
    }
  }

  // lanes L and L+16 hold partial sums for the same frequency column
  accRe += __shfl_xor(accRe, 16, 32);
  accIm += __shfl_xor(accIm, 16, 32);

  if (!hi) {
    const int idx = ((b * FX + x) * FY2 + y) * 2;
    out[idx + 0] = -16384.0f * accRe;   // F = -F; F *= RES^J
    out[idx + 1] = -16384.0f * accIm;
  }
}

// ---------------------------------------------------------------------------
// Kernel 3: DC component. out[b,0,0,:,:] = sum(C*D) * 8192 (both components)
// ---------------------------------------------------------------------------
__global__ void ddsl_dc(const float* __restrict__ elem, float* __restrict__ out) {
  __shared__ float red[256];
  const int b = blockIdx.x;
  float s = 0.0f;
  for (int e = threadIdx.x; e < NVELEM; e += 256)
    s += elem[(b * NVELEM + e) * 5 + 4];
  red[threadIdx.x] = s;
  __syncthreads();
  for (int off = 128; off > 0; off >>= 1) {
    if (threadIdx.x < off) red[threadIdx.x] += red[threadIdx.x + off];
    __syncthreads();
  }
  if (threadIdx.x == 0) {
    // F(0,0) = -(-sum(CD)/2) * 16384 = sum(CD) * 8192, set on re and im
    const float v = red[0] * 8192.0f;
    const int idx = (b * FX * FY2) * 2;
    out[idx + 0] = v;
    out[idx + 1] = v;
  }
}

// ---------------------------------------------------------------------------
extern "C" void kernel_launch(void* const* d_in, const int* in_sizes, int n_in,
                              void* d_out, int out_size, void* d_ws, size_t ws_size,
                              hipStream_t stream) {
  const float* V = (const float*)d_in[0];  // (B, NV, 2) f32
  const int*   E = (const int*)d_in[1];    // (B, NV, 2) i32
  const float* D = (const float*)d_in[2];  // (B, NV, 1) f32
  float* out  = (float*)d_out;             // (B, 128, 65, 1, 2) f32
  float* elem = (float*)d_ws;              // B*NV*5 floats = 20 KB scratch

  ddsl_prep<<<(BATCH * NVELEM + 255) / 256, 256, 0, stream>>>(V, E, D, elem);
  ddsl_main<<<BATCH * BLOCKS_PER_B, 256, 0, stream>>>(elem, out);
  ddsl_dc<<<BATCH, 256, 0, stream>>>(elem, out);
}